// IjspeertActor_14663018349111
// MI455X (gfx1250) — compile-verified
//
#include <hip/hip_runtime.h>
#include <math.h>

#define NOSC 5
#define NSTEPS 10

typedef __attribute__((ext_vector_type(4))) float v4f;

__device__ __forceinline__ float softplus_f(float x) {
    // log(1 + exp(x)), stable for large x
    return (x > 20.0f) ? x : log1pf(expf(x));
}

// O(1) setup: the RK4 trajectory is batch-invariant (state starts at zero and
// the dynamics never touch per-batch data), so integrate ONCE.
// Lane-parallel layout: lane i (i < 5) owns component i of (r, r_dot, theta);
// cross-component terms come from wave32 shuffles. Dynamics per stage:
//   rdd_i = a_i*(a_i/4*(mu_i - r_i) - rd_i)
//   td_i  = f_i + sum_j W[i][j]*sin(th_j - th_i - PHI[i][j])*r_j
__global__ void ijspeert_setup_kernel(const float* __restrict__ dt_p,
                                      const float* __restrict__ mu_p_g,
                                      const float* __restrict__ a_g,
                                      const float* __restrict__ w_g,
                                      const float* __restrict__ phi_g,
                                      const float* __restrict__ freq_g,
                                      const float* __restrict__ amp_g,
                                      const float* __restrict__ pha_g,
                                      const float* __restrict__ std_g,
                                      float* __restrict__ ws) {
    const int lane = threadIdx.x;                 // launched as <<<1, 32>>>
    const int i = (lane < NOSC) ? lane : 0;       // clamp: lanes 5..31 shadow lane 0
                                                  // (no divergence -> shuffles valid)
    const float dt = dt_p[0];
    const float h2 = 0.5f * dt;
    const float h6 = dt * (1.0f / 6.0f);

    const float mu_i = mu_p_g[i];
    const float a_i  = a_g[i];
    const float fr_i = freq_g[i];
    float Wi[NOSC], PHIi[NOSC];
#pragma unroll
    for (int j = 0; j < NOSC; ++j) {
        Wi[j]   = w_g[i * NOSC + j];
        PHIi[j] = phi_g[i * NOSC + j];
    }

    float r = 0.0f, rd = 0.0f, th = 0.0f;

    // one dynamics evaluation at state (r_s, rd_s, th_s): 5 native sins / lane
    auto dyn = [&](float r_s, float rd_s, float th_s, float& rdd, float& td) {
        rdd = a_i * (a_i * 0.25f * (mu_i - r_s) - rd_s);
        float c = 0.0f;
#pragma unroll
        for (int j = 0; j < NOSC; ++j) {
            const float th_j = __shfl(th_s, j, 32);
            const float r_j  = __shfl(r_s,  j, 32);
            c += Wi[j] * __sinf(th_j - th_s - PHIi[j]) * r_j;
        }
        td = fr_i + c;
    };

#pragma unroll 1
    for (int s = 0; s < NSTEPS; ++s) {
        float rdd1, td1, rdd2, td2, rdd3, td3, rdd4, td4;
        dyn(r, rd, th, rdd1, td1);
        const float r2  = r  + h2 * rd;           // k1_r  = rd
        const float rd2 = rd + h2 * rdd1;         // rd2 == k2_r
        const float th2 = th + h2 * td1;
        dyn(r2, rd2, th2, rdd2, td2);
        const float r3  = r  + h2 * rd2;
        const float rd3 = rd + h2 * rdd2;         // rd3 == k3_r
        const float th3 = th + h2 * td2;
        dyn(r3, rd3, th3, rdd3, td3);
        const float r4  = r  + dt * rd3;
        const float rd4 = rd + dt * rdd3;         // rd4 == k4_r
        const float th4 = th + dt * td3;
        dyn(r4, rd4, th4, rdd4, td4);
        r  += h6 * (rd   + 2.0f * rd2  + 2.0f * rd3  + rd4);
        rd += h6 * (rdd1 + 2.0f * rdd2 + 2.0f * rdd3 + rdd4);
        th += h6 * (td1  + 2.0f * td2  + 2.0f * td3  + td4);
    }

    if (lane < NOSC) {
        // RANGE = [pi/2, pi, pi/2, pi, pi/2] -> odd lanes get pi
        const float range_i = (lane & 1) ? 3.14159265358979323846f
                                         : 1.57079632679489661923f;
        const float two_pi = 6.283185307179586f;
        ws[lane]            = tanhf(amp_g[lane]) * range_i;      // amplitude
        ws[NOSC + lane]     = two_pi * softplus_f(freq_g[lane]); // angular freq
        ws[2 * NOSC + lane] = softplus_f(pha_g[lane]);           // phase
        ws[3 * NOSC + lane] = std_g[lane];
        ws[4 * NOSC + lane] = r;
        ws[5 * NOSC + lane] = th;
    }
}

// Streaming kernel: each thread owns 4 batch elements = 20 floats per plane.
// tid*20 floats = tid*80 bytes -> 16B aligned, and tid*20 % 5 == 0 so every
// chunk starts at oscillator 0: the std/r/theta planes are a constant 20-float
// register pattern -> pure aligned global_store_b128 streams. 88 MB total
// traffic -> ~3.8 us HBM floor (and the whole working set fits the 192 MB L2,
// so default RT stores may run replays at L2 speed; deliberately NOT using NT).
__global__ __launch_bounds__(256)
void ijspeert_stream_kernel(const float* __restrict__ t,
                            const float* __restrict__ ws,
                            float* __restrict__ out, int B) {
    const int tid = blockIdx.x * blockDim.x + threadIdx.x;
    const long long P = (long long)B * NOSC;

    float amp[NOSC], w2[NOSC], ph[NOSC], c_std[NOSC], c_r[NOSC], c_th[NOSC];
#pragma unroll
    for (int i = 0; i < NOSC; ++i) {
        amp[i]   = ws[i];
        w2[i]    = ws[NOSC + i];
        ph[i]    = ws[2 * NOSC + i];
        c_std[i] = ws[3 * NOSC + i];
        c_r[i]   = ws[4 * NOSC + i];
        c_th[i]  = ws[5 * NOSC + i];
    }

    const int b0 = tid * 4;
    if (b0 + 4 <= B) {
        const v4f tv = *(const v4f*)(t + b0);   // 16B-aligned b128 load
        float mu[20];
#pragma unroll
        for (int k = 0; k < 4; ++k) {
            const float tb = tv[k];
#pragma unroll
            for (int i = 0; i < NOSC; ++i)
                mu[k * NOSC + i] = amp[i] * __sinf(fmaf(w2[i], tb, ph[i]));
        }
        const size_t base = (size_t)tid * 20;
        v4f* o0 = (v4f*)(out + base);                      // mu
        v4f* o1 = (v4f*)(out + (size_t)P + base);          // std
        v4f* o2 = (v4f*)(out + (size_t)(2 * P) + base);    // r
        v4f* o3 = (v4f*)(out + (size_t)(3 * P) + base);    // theta
#pragma unroll
        for (int q = 0; q < 5; ++q) {
            v4f m, s, rr, tt;
#pragma unroll
            for (int e = 0; e < 4; ++e) {
                const int idx = q * 4 + e;          // 0..19 within chunk
                m[e]  = mu[idx];
                s[e]  = c_std[idx % NOSC];
                rr[e] = c_r[idx % NOSC];
                tt[e] = c_th[idx % NOSC];
            }
            o0[q] = m; o1[q] = s; o2[q] = rr; o3[q] = tt;
        }
    } else if (b0 < B) {
        // tail (not reached for B = 2^20, kept for safety)
        for (int k = 0; k < 4 && b0 + k < B; ++k) {
            const int b = b0 + k;
            const float tb = t[b];
            for (int i = 0; i < NOSC; ++i) {
                const size_t o = (size_t)b * NOSC + i;
                out[o]                     = amp[i] * __sinf(fmaf(w2[i], tb, ph[i]));
                out[(size_t)P + o]         = c_std[i];
                out[(size_t)(2 * P) + o]   = c_r[i];
                out[(size_t)(3 * P) + o]   = c_th[i];
            }
        }
    }
}

extern "C" void kernel_launch(void* const* d_in, const int* in_sizes, int n_in,
                              void* d_out, int out_size, void* d_ws, size_t ws_size,
                              hipStream_t stream) {
    // setup_inputs order:
    // 0: obs (unused)  1: t  2: dt  3: mu_p  4: a  5: w  6: phi
    // 7: frequencies   8: amplitudes  9: phases  10: std
    const float* t_in  = (const float*)d_in[1];
    const float* dt_in = (const float*)d_in[2];
    const float* mu_p  = (const float*)d_in[3];
    const float* a_in  = (const float*)d_in[4];
    const float* w_in  = (const float*)d_in[5];
    const float* phi   = (const float*)d_in[6];
    const float* freq  = (const float*)d_in[7];
    const float* ampl  = (const float*)d_in[8];
    const float* phas  = (const float*)d_in[9];
    const float* stdv  = (const float*)d_in[10];
    float* out = (float*)d_out;
    float* ws  = (float*)d_ws;

    const int B = in_sizes[1];

    // O(1) batch-invariant RK4 (lane-parallel, one wave) -> 30 floats in ws
    ijspeert_setup_kernel<<<1, 32, 0, stream>>>(dt_in, mu_p, a_in, w_in, phi,
                                                freq, ampl, phas, stdv, ws);

    // Bandwidth-bound broadcast/sin stream: 4 batch elements per thread
    const int threads_total = (B + 3) / 4;
    const int block = 256;                          // 8 wave32 per block
    const int grid = (threads_total + block - 1) / block;
    ijspeert_stream_kernel<<<grid, block, 0, stream>>>(t_in, ws, out, B);
}